// RC_STML_91285234909293
// MI455X (gfx1250) — compile-verified
//
#include <hip/hip_runtime.h>
#include <math.h>
#include <stdint.h>

#define N 4096
#define D 512
#define TOPKN 10
#define TOPK_HALF 5
#define MAXHAT (TOPKN * TOPK_HALF) /* 50 */

#define KC 32      /* k-chunk staged in LDS */
#define KCP 40     /* padded row stride (floats): 160B, 16B-aligned, spreads banks */

typedef float v2f __attribute__((ext_vector_type(2)));
typedef float v8f __attribute__((ext_vector_type(8)));

#if defined(__HIP_DEVICE_COMPILE__) && __has_builtin(__builtin_amdgcn_global_load_async_to_lds_b128)
#define HAVE_ASYNC_LDS 1
typedef int v4i_t __attribute__((ext_vector_type(4)));
typedef __attribute__((address_space(1))) v4i_t* gp_v4i;   // global int4*
typedef __attribute__((address_space(3))) v4i_t* lp_v4i;   // LDS int4*
#else
#define HAVE_ASYNC_LDS 0
#endif

__device__ __forceinline__ void wait_async_copies() {
#if HAVE_ASYNC_LDS
#if __has_builtin(__builtin_amdgcn_s_wait_asynccnt)
    __builtin_amdgcn_s_wait_asynccnt(0);
#else
    asm volatile("s_wait_asynccnt 0x0" ::: "memory");
#endif
#endif
}

// Stage one KC-wide chunk of A-rows and B-rows into LDS (cooperative, 256 threads).
__device__ __forceinline__ void copy_chunk(const float* __restrict__ gA,
                                           const float* __restrict__ gB,
                                           float* __restrict__ ldsA,
                                           float* __restrict__ ldsB,
                                           int kbase, int tid) {
#pragma unroll
    for (int i = 0; i < 4; ++i) {
        const int f = tid + 256 * i;       // 1024 float4 per matrix chunk
        const int row = f >> 3;            // 8 float4 per 32-float row
        const int c4 = (f & 7) * 4;
        const float* ga = gA + (size_t)row * D + kbase + c4;
        const float* gb = gB + (size_t)row * D + kbase + c4;
        float* la = ldsA + row * KCP + c4;
        float* lb = ldsB + row * KCP + c4;
#if HAVE_ASYNC_LDS
        __builtin_amdgcn_global_load_async_to_lds_b128((gp_v4i)(uintptr_t)ga,
                                                       (lp_v4i)(uintptr_t)la, 0, 0);
        __builtin_amdgcn_global_load_async_to_lds_b128((gp_v4i)(uintptr_t)gb,
                                                       (lp_v4i)(uintptr_t)lb, 0, 0);
#else
        *(float4*)la = *(const float4*)ga;
        *(float4*)lb = *(const float4*)gb;
#endif
    }
}

// ---------------- row sum-of-squares (for s_emb) ----------------
__global__ void rownorm_kernel(const float* __restrict__ X, float* __restrict__ ns) {
    __shared__ float sbuf[256];
    const int row = blockIdx.x;
    const float* p = X + (size_t)row * D;
    float acc = 0.f;
    for (int k = threadIdx.x; k < D; k += 256) { float v = p[k]; acc += v * v; }
    sbuf[threadIdx.x] = acc; __syncthreads();
    for (int o = 128; o > 0; o >>= 1) {
        if (threadIdx.x < o) sbuf[threadIdx.x] += sbuf[threadIdx.x + o];
        __syncthreads();
    }
    if (threadIdx.x == 0) ns[row] = sbuf[0];
}

// ---------------- normalize t rows, emit t_norm and its row norms^2 ----------------
__global__ void tnorm_kernel(const float* __restrict__ T, float* __restrict__ Tn,
                             float* __restrict__ nt) {
    __shared__ float sbuf[256];
    const int row = blockIdx.x;
    const float* p = T + (size_t)row * D;
    float acc = 0.f;
    for (int k = threadIdx.x; k < D; k += 256) { float v = p[k]; acc += v * v; }
    sbuf[threadIdx.x] = acc; __syncthreads();
    for (int o = 128; o > 0; o >>= 1) {
        if (threadIdx.x < o) sbuf[threadIdx.x] += sbuf[threadIdx.x + o];
        __syncthreads();
    }
    const float ss = sbuf[0];
    float nrm = sqrtf(ss); nrm = nrm > 1e-12f ? nrm : 1e-12f;
    const float inv = 1.0f / nrm;
    float* q = Tn + (size_t)row * D;
    for (int k = threadIdx.x; k < D; k += 256) q[k] = p[k] * inv;
    if (threadIdx.x == 0) nt[row] = ss * inv * inv;
}

// ---------------- WMMA Gram -> distance/affinity, LDS double-buffered ----------------
// Block: 256 thr (8 waves), output tile 128x128; wave w -> 16-row strip x 128 cols.
// mode 0: out = sqrt(relu(ni+nj-2G))   (S_dist)
// mode 1: out = exp(-relu(ni+nj-2G))   (W_P, SIGMA=1)
__global__ void __launch_bounds__(256)
gram_kernel(const float* __restrict__ X, const float* __restrict__ nrm,
            float* __restrict__ out, int mode) {
    __shared__ float lA[2][128 * KCP];
    __shared__ float lB[2][128 * KCP];

    const int tid  = threadIdx.x;
    const int lane = tid & 31;
    const int wave = tid >> 5;
    const int rl   = lane & 15;
    const int koff = (lane >> 4) * 2;   // fp32 16x4 A/B frag: lanes>=16 hold K+2..K+3

    const float* gA = X + (size_t)(blockIdx.x * 128) * D;  // A rows
    const float* gB = X + (size_t)(blockIdx.y * 128) * D;  // B rows (= output cols)

    v8f acc[8];
#pragma unroll
    for (int q = 0; q < 8; ++q) acc[q] = (v8f){};

    // prologue: stage chunk 0
    copy_chunk(gA, gB, &lA[0][0], &lB[0][0], 0, tid);
    wait_async_copies();
    __syncthreads();

    const int nchunks = D / KC;   // 16
    for (int c = 0; c < nchunks; ++c) {
        const int sel = c & 1;
        if (c + 1 < nchunks)
            copy_chunk(gA, gB, &lA[sel ^ 1][0], &lB[sel ^ 1][0], (c + 1) * KC, tid);

        const float* la = &lA[sel][(wave * 16 + rl) * KCP + koff];
        const float* lb = &lB[sel][rl * KCP + koff];
#pragma unroll
        for (int kk = 0; kk < KC; kk += 4) {
            v2f a = *(const v2f*)(la + kk);
#pragma unroll
            for (int q = 0; q < 8; ++q) {
                v2f b = *(const v2f*)(lb + q * 16 * KCP + kk);
                acc[q] = __builtin_amdgcn_wmma_f32_16x16x4_f32(
                    false, a, false, b, (short)0, acc[q], false, false);
            }
        }

        if (c + 1 < nchunks) wait_async_copies();
        __syncthreads();
    }

    // C layout: VGPR v, lanes 0-15 -> row v, lanes 16-31 -> row v+8; col = lane&15
    const int row0 = blockIdx.x * 128 + wave * 16 + (lane >> 4) * 8;
    const int cb   = blockIdx.y * 128;
#pragma unroll
    for (int v = 0; v < 8; ++v) {
        const int i = row0 + v;
        const float ni = nrm[i];
#pragma unroll
        for (int q = 0; q < 8; ++q) {
            const int j = cb + q * 16 + rl;
            float d2 = ni + nrm[j] - 2.0f * acc[q][v];
            d2 = d2 > 0.f ? d2 : 0.f;
            out[(size_t)i * N + j] = (mode == 0) ? sqrtf(d2) : expf(-d2);
        }
    }
}

// ---------------- row mean of S_dist ----------------
__global__ void rowmean_kernel(const float* __restrict__ Sd, float* __restrict__ mean) {
    __shared__ float sbuf[256];
    const int row = blockIdx.x;
    const float* p = Sd + (size_t)row * N;
    float acc = 0.f;
    for (int j = threadIdx.x; j < N; j += 256) acc += p[j];
    sbuf[threadIdx.x] = acc; __syncthreads();
    for (int o = 128; o > 0; o >>= 1) {
        if (threadIdx.x < o) sbuf[threadIdx.x] += sbuf[threadIdx.x + o];
        __syncthreads();
    }
    if (threadIdx.x == 0) mean[row] = sbuf[0] * (1.0f / (float)N);
}

// ---------------- stable top-10 per row of W_P_copy ----------------
__global__ void topk_kernel(const float* __restrict__ WP, const int* __restrict__ idx,
                            int* __restrict__ tk) {
    __shared__ float lv[128 * TOPKN];
    __shared__ int   li[128 * TOPKN];
    const int row = blockIdx.x;
    const int t = threadIdx.x;
    const int my = idx[row];
    const float* wr = WP + (size_t)row * N;

    float bv[TOPKN]; int bi[TOPKN];
#pragma unroll
    for (int e = 0; e < TOPKN; ++e) { bv[e] = -3.4e38f; bi[e] = 0x7fffffff; }
    for (int j = t; j < N; j += 128) {
        const float v = (idx[j] == my) ? 1.0f : wr[j];
        if (v > bv[TOPKN-1] || (v == bv[TOPKN-1] && j < bi[TOPKN-1])) {
            int p = TOPKN - 1;
            while (p > 0 && (v > bv[p-1] || (v == bv[p-1] && j < bi[p-1]))) {
                bv[p] = bv[p-1]; bi[p] = bi[p-1]; --p;
            }
            bv[p] = v; bi[p] = j;
        }
    }
#pragma unroll
    for (int e = 0; e < TOPKN; ++e) { lv[t*TOPKN+e] = bv[e]; li[t*TOPKN+e] = bi[e]; }
    __syncthreads();
    if (t == 0) {
        float fv[TOPKN]; int fi[TOPKN];
#pragma unroll
        for (int e = 0; e < TOPKN; ++e) { fv[e] = -3.4e38f; fi[e] = 0x7fffffff; }
        for (int q = 0; q < 128 * TOPKN; ++q) {
            const float v = lv[q]; const int j = li[q];
            if (j == 0x7fffffff) continue;
            if (v > fv[TOPKN-1] || (v == fv[TOPKN-1] && j < fi[TOPKN-1])) {
                int p = TOPKN - 1;
                while (p > 0 && (v > fv[p-1] || (v == fv[p-1] && j < fi[p-1]))) {
                    fv[p] = fv[p-1]; fi[p] = fi[p-1]; --p;
                }
                fv[p] = v; fi[p] = j;
            }
        }
        for (int e = 0; e < TOPKN; ++e) tk[row*TOPKN+e] = fi[e];
    }
}

// ---------------- mutual kNN: V row lists (<=10 nonzeros/row) ----------------
__global__ void mutual_kernel(const int* __restrict__ tk, int* __restrict__ vlist,
                              int* __restrict__ vcount) {
    const int i = blockIdx.x * blockDim.x + threadIdx.x;
    if (i >= N) return;
    int cnt = 0;
    for (int e = 0; e < TOPKN; ++e) {
        const int j = tk[i*TOPKN+e];
        bool m = false;
        for (int f = 0; f < TOPKN; ++f) m |= (tk[j*TOPKN+f] == i);
        if (m) { vlist[i*TOPKN+cnt] = j; ++cnt; }
    }
    vcount[i] = cnt;
}

// ---------------- W_C_tilda values aligned with vlist ----------------
__global__ void wct_kernel(const int* __restrict__ vlist, const int* __restrict__ vcount,
                           float* __restrict__ wct) {
    const int i = blockIdx.x * blockDim.x + threadIdx.x;
    if (i >= N) return;
    const int ci = vcount[i];
    const float invdeg = 1.0f / fmaxf((float)ci, 1.0f);
    for (int e = 0; e < ci; ++e) {
        const int j = vlist[i*TOPKN+e];
        const int cj = vcount[j];
        int co = 0;
        for (int a = 0; a < ci; ++a) {
            const int ka = vlist[i*TOPKN+a];
            for (int b = 0; b < cj; ++b) co += (vlist[j*TOPKN+b] == ka) ? 1 : 0;
        }
        wct[i*TOPKN+e] = (float)co * invdeg;
    }
}

// ---------------- sparse W_C_hat rows (<=50 entries each) ----------------
__global__ void hat_kernel(const int* __restrict__ tk, const int* __restrict__ vlist,
                           const int* __restrict__ vcount, const float* __restrict__ wct,
                           int* __restrict__ hcols, float* __restrict__ hvals,
                           int* __restrict__ hcnt) {
    const int i = blockIdx.x * blockDim.x + threadIdx.x;
    if (i >= N) return;
    int cnt = 0;
    for (int h = 0; h < TOPK_HALF; ++h) {
        const int m = tk[i*TOPKN+h];
        const int cm = vcount[m];
        for (int e = 0; e < cm; ++e) {
            hcols[i*MAXHAT+cnt] = vlist[m*TOPKN+e];
            hvals[i*MAXHAT+cnt] = wct[m*TOPKN+e] * (1.0f / (float)TOPK_HALF);
            ++cnt;
        }
    }
    hcnt[i] = cnt;
}

// ---------------- dense loss term (W_P half) ----------------
__global__ void dense_loss_kernel(const float* __restrict__ Sd, const float* __restrict__ WP,
                                  const float* __restrict__ mean, float* __restrict__ part) {
    __shared__ float sbuf[256];
    const int i = blockIdx.x;
    const float inv = 1.0f / mean[i];
    const float* sr = Sd + (size_t)i * N;
    const float* wr = WP + (size_t)i * N;
    float acc = 0.f;
    for (int j = threadIdx.x; j < N; j += 256) {
        if (j == i) continue;
        const float S = sr[j] * inv;
        float p = 1.0f - S; p = p > 0.f ? p : 0.f;
        const float rp2 = p * p;
        acc += rp2 + 0.5f * wr[j] * (S * S - rp2);
    }
    sbuf[threadIdx.x] = acc; __syncthreads();
    for (int o = 128; o > 0; o >>= 1) {
        if (threadIdx.x < o) sbuf[threadIdx.x] += sbuf[threadIdx.x + o];
        __syncthreads();
    }
    if (threadIdx.x == 0) part[i] = sbuf[0];
}

// ---------------- sparse loss correction (W_C half) ----------------
__global__ void sparse_loss_kernel(const float* __restrict__ Sd, const float* __restrict__ mean,
                                   const int* __restrict__ hcols, const float* __restrict__ hvals,
                                   const int* __restrict__ hcnt, float* __restrict__ part) {
    __shared__ float sbuf[64];
    const int i = blockIdx.x;
    const int cnt = hcnt[i];
    const int t = threadIdx.x;
    float acc = 0.f;
    if (t < cnt) {
        const int j = hcols[i*MAXHAT+t];
        if (j != i) {
            const float v = hvals[i*MAXHAT+t];
            const float Sij = Sd[(size_t)i*N+j] / mean[i];
            const float Sji = Sd[(size_t)j*N+i] / mean[j];
            float pij = 1.f - Sij; pij = pij > 0.f ? pij : 0.f;
            float pji = 1.f - Sji; pji = pji > 0.f ? pji : 0.f;
            acc = 0.25f * v * ((Sij*Sij - pij*pij) + (Sji*Sji - pji*pji));
        }
    }
    sbuf[t] = acc; __syncthreads();
    for (int o = 32; o > 0; o >>= 1) {
        if (t < o) sbuf[t] += sbuf[t + o];
        __syncthreads();
    }
    if (t == 0) part[i] = sbuf[0];
}

// ---------------- final scalar reduce ----------------
__global__ void final_kernel(const float* __restrict__ dpart, const float* __restrict__ spart,
                             float* __restrict__ out) {
    __shared__ float sbuf[256];
    float acc = 0.f;
    for (int q = threadIdx.x; q < N; q += 256) acc += dpart[q] + spart[q];
    sbuf[threadIdx.x] = acc; __syncthreads();
    for (int o = 128; o > 0; o >>= 1) {
        if (threadIdx.x < o) sbuf[threadIdx.x] += sbuf[threadIdx.x + o];
        __syncthreads();
    }
    if (threadIdx.x == 0) out[0] = sbuf[0] / ((float)N * (float)(N - 1));
}

extern "C" void kernel_launch(void* const* d_in, const int* in_sizes, int n_in,
                              void* d_out, int out_size, void* d_ws, size_t ws_size,
                              hipStream_t stream) {
    (void)in_sizes; (void)n_in; (void)out_size; (void)ws_size;
    const float* s_emb = (const float*)d_in[0];
    const float* t_emb = (const float*)d_in[1];
    const int*   idx   = (const int*)d_in[2];
    float* out = (float*)d_out;

    char* base = (char*)d_ws;
    size_t off = 0;
    auto alloc = [&](size_t bytes) -> void* {
        void* p = base + off;
        off = (off + bytes + 255) & ~(size_t)255;
        return p;
    };
    float* t_norm = (float*)alloc((size_t)N * D * 4);
    float* Sd     = (float*)alloc((size_t)N * N * 4);
    float* WP     = (float*)alloc((size_t)N * N * 4);
    float* ns     = (float*)alloc((size_t)N * 4);
    float* nt     = (float*)alloc((size_t)N * 4);
    float* mean   = (float*)alloc((size_t)N * 4);
    int*   tk     = (int*)alloc((size_t)N * TOPKN * 4);
    int*   vlist  = (int*)alloc((size_t)N * TOPKN * 4);
    int*   vcount = (int*)alloc((size_t)N * 4);
    float* wct    = (float*)alloc((size_t)N * TOPKN * 4);
    int*   hcols  = (int*)alloc((size_t)N * MAXHAT * 4);
    float* hvals  = (float*)alloc((size_t)N * MAXHAT * 4);
    int*   hcnt   = (int*)alloc((size_t)N * 4);
    float* dpart  = (float*)alloc((size_t)N * 4);
    float* spart  = (float*)alloc((size_t)N * 4);

    rownorm_kernel<<<N, 256, 0, stream>>>(s_emb, ns);
    tnorm_kernel<<<N, 256, 0, stream>>>(t_emb, t_norm, nt);

    dim3 ggrid(N / 128, N / 128);
    gram_kernel<<<ggrid, 256, 0, stream>>>(s_emb, ns, Sd, 0);
    gram_kernel<<<ggrid, 256, 0, stream>>>(t_norm, nt, WP, 1);

    rowmean_kernel<<<N, 256, 0, stream>>>(Sd, mean);
    topk_kernel<<<N, 128, 0, stream>>>(WP, idx, tk);
    mutual_kernel<<<N / 256, 256, 0, stream>>>(tk, vlist, vcount);
    wct_kernel<<<N / 256, 256, 0, stream>>>(vlist, vcount, wct);
    hat_kernel<<<N / 256, 256, 0, stream>>>(tk, vlist, vcount, wct, hcols, hvals, hcnt);
    dense_loss_kernel<<<N, 256, 0, stream>>>(Sd, WP, mean, dpart);
    sparse_loss_kernel<<<N, 64, 0, stream>>>(Sd, mean, hcols, hvals, hcnt, spart);
    final_kernel<<<1, 256, 0, stream>>>(dpart, spart, out);
}